// GraphVAE_64914135711787
// MI455X (gfx1250) — compile-verified
//
#include <hip/hip_runtime.h>
#include <hip/hip_bf16.h>
#include <math.h>

#define N_NODES 20000
#define N_EDGES 320000
#define HD      128
#define NHEAD   4
#define NGRAPH  8
#define NCLASS  10

typedef float v2f __attribute__((ext_vector_type(2)));
typedef float v8f __attribute__((ext_vector_type(8)));

#define XS_STRIDE 132   // 16B-aligned rows (132*4 % 16 == 0), bank-conflict-free A reads

// -----------------------------------------------------------------------------
// Y[nrows x 128] = X[nrows x 128] @ W[128 x 128] + bias, exact f32 via
// V_WMMA_F32_16X16X4_F32. Block = 8 waves; wave w owns columns [16w,16w+16).
// A tile (16 rows x 128 K) staged in LDS, shared by all 8 waves.
// -----------------------------------------------------------------------------
__global__ __launch_bounds__(256) void gemm128_wmma_f32(
    const float* __restrict__ X, const float* __restrict__ W,
    const float* __restrict__ bias, float* __restrict__ Y, int nrows)
{
  __shared__ float xs[16 * XS_STRIDE];
  const int tid  = threadIdx.x;
  const int lane = tid & 31;
  const int wave = tid >> 5;                 // 0..7 -> column tile
  const int rowBase = blockIdx.x * 16;

  // Cooperative coalesced float4 stage of the 16x128 X tile.
  for (int i = tid; i < 16 * 32; i += 256) {
    const int r  = i >> 5;
    const int c4 = i & 31;
    float4 val = {0.f, 0.f, 0.f, 0.f};
    const int gr = rowBase + r;
    if (gr < nrows) val = *(const float4*)(X + (size_t)gr * HD + c4 * 4);
    *(float4*)(&xs[r * XS_STRIDE + c4 * 4]) = val;
  }
  __syncthreads();

  const int colBase = wave * 16;
  const int col     = colBase + (lane & 15);
  const int rsel    = lane & 15;             // A: lanes 0-15 / 16-31 both M=0..15
  const int khalf   = (lane >> 4) * 2;       // A/B: K pair 0-1 (lo half) or 2-3 (hi half)

  v8f acc = {0.f, 0.f, 0.f, 0.f, 0.f, 0.f, 0.f, 0.f};
  #pragma unroll
  for (int k0 = 0; k0 < HD; k0 += 4) {
    v2f a = *(const v2f*)(&xs[rsel * XS_STRIDE + k0 + khalf]);
    v2f b;
    b.x = W[(size_t)(k0 + khalf + 0) * HD + col];
    b.y = W[(size_t)(k0 + khalf + 1) * HD + col];
    acc = __builtin_amdgcn_wmma_f32_16x16x4_f32(false, a, false, b,
                                                (short)0, acc, false, false);
  }

  // C/D layout: VGPR i -> M = i + 8*(lane>=16), N = lane&15 (+colBase)
  const float bv   = bias[col];
  const int   rtop = (lane >> 4) * 8;
  #pragma unroll
  for (int i = 0; i < 8; ++i) {
    const int gr = rowBase + rtop + i;
    if (gr < nrows) Y[(size_t)gr * HD + col] = acc[i] + bv;
  }
}

// -----------------------------------------------------------------------------
__global__ void fill_kernel(float* __restrict__ p, float v, int n)
{
  const int i = blockIdx.x * blockDim.x + threadIdx.x;
  if (i < n) p[i] = v;
}

// -----------------------------------------------------------------------------
// One wave per edge: logit[e,h] = dot(q[dst], k[src] + attr*We + be)/sqrt(32),
// plus float segment-max via int-punned atomics (m initialized to -inf).
// -----------------------------------------------------------------------------
__global__ __launch_bounds__(256) void edge_logits_kernel(
    const float* __restrict__ q, const float* __restrict__ k,
    const int* __restrict__ src, const int* __restrict__ dst,
    const float* __restrict__ attr, const float* __restrict__ We,
    const float* __restrict__ be,
    float* __restrict__ lgOut, float* __restrict__ mb, int nE)
{
  const int wid  = (blockIdx.x * blockDim.x + threadIdx.x) >> 5;
  const int lane = threadIdx.x & 31;
  if (wid >= nE) return;
  const int   sj = src[wid], di = dst[wid];
  const float av = attr[wid];
  const float4 q4 = *(const float4*)(q + (size_t)di * HD + lane * 4);
  const float4 k4 = *(const float4*)(k + (size_t)sj * HD + lane * 4);
  const float4 w4 = *(const float4*)(We + lane * 4);
  const float4 b4 = *(const float4*)(be + lane * 4);
  float p = q4.x * (k4.x + av * w4.x + b4.x)
          + q4.y * (k4.y + av * w4.y + b4.y)
          + q4.z * (k4.z + av * w4.z + b4.z)
          + q4.w * (k4.w + av * w4.w + b4.w);
  // reduce 8 lanes per head (head h = lane/8, D=32 -> 4 floats/lane)
  p += __shfl_xor(p, 1, 32);
  p += __shfl_xor(p, 2, 32);
  p += __shfl_xor(p, 4, 32);
  if ((lane & 7) == 0) {
    const int   h  = lane >> 3;
    const float lg = p * 0.17677669529663687f;   // 1/sqrt(32)
    lgOut[(size_t)wid * NHEAD + h] = lg;
    float* mp = &mb[(size_t)di * NHEAD + h];
    if (lg >= 0.f) atomicMax((int*)mp, __float_as_int(lg));
    else           atomicMin((unsigned int*)mp, __float_as_uint(lg));
  }
}

// -----------------------------------------------------------------------------
// ex = exp(logit - m[dst]); segment-sum into s[dst]; ex overwrites logit buffer.
// -----------------------------------------------------------------------------
__global__ __launch_bounds__(256) void edge_exp_kernel(
    const int* __restrict__ dst, float* __restrict__ lg,
    const float* __restrict__ mb, float* __restrict__ sb, int nEH)
{
  const int t = blockIdx.x * blockDim.x + threadIdx.x;
  if (t >= nEH) return;
  const int e = t >> 2, h = t & 3;
  const int d = dst[e];
  const float ex = __expf(lg[t] - mb[(size_t)d * NHEAD + h]);
  lg[t] = ex;
  atomicAdd(&sb[(size_t)d * NHEAD + h], ex);
}

// -----------------------------------------------------------------------------
// One wave per edge: alpha = ex/(s[dst]+1e-16) (written back to the a-output
// buffer), out[dst] += alpha * (v[src] + attr*We + be). outBuf is pre-seeded
// with the skip projection x@Ws + bs.
// -----------------------------------------------------------------------------
__global__ __launch_bounds__(256) void edge_scatter_kernel(
    const float* __restrict__ v, const int* __restrict__ src,
    const int* __restrict__ dst, const float* __restrict__ attr,
    const float* __restrict__ We, const float* __restrict__ be,
    float* __restrict__ exAlpha, const float* __restrict__ sb,
    float* __restrict__ outBuf, int nE)
{
  const int wid  = (blockIdx.x * blockDim.x + threadIdx.x) >> 5;
  const int lane = threadIdx.x & 31;
  if (wid >= nE) return;
  const int   sj = src[wid], di = dst[wid];
  const float av = attr[wid];
  const int   h  = lane >> 3;
  const float ex    = exAlpha[(size_t)wid * NHEAD + h];           // load (all lanes)
  const float alpha = ex / (sb[(size_t)di * NHEAD + h] + 1e-16f);
  if ((lane & 7) == 0) exAlpha[(size_t)wid * NHEAD + h] = alpha;  // store after load
  const float4 v4 = *(const float4*)(v + (size_t)sj * HD + lane * 4);
  const float4 w4 = *(const float4*)(We + lane * 4);
  const float4 b4 = *(const float4*)(be + lane * 4);
  float* o = outBuf + (size_t)di * HD + lane * 4;
  atomicAdd(o + 0, alpha * (v4.x + av * w4.x + b4.x));
  atomicAdd(o + 1, alpha * (v4.y + av * w4.y + b4.y));
  atomicAdd(o + 2, alpha * (v4.z + av * w4.z + b4.z));
  atomicAdd(o + 3, alpha * (v4.w + av * w4.w + b4.w));
}

// -----------------------------------------------------------------------------
// relu(LayerNorm(row)) — one wave per 128-wide row, full-wave shuffle reduce.
// -----------------------------------------------------------------------------
__global__ __launch_bounds__(256) void ln_relu_kernel(
    const float* __restrict__ X, const float* __restrict__ gw,
    const float* __restrict__ bw, float* __restrict__ Y, int nrows)
{
  const int wid  = (blockIdx.x * blockDim.x + threadIdx.x) >> 5;
  const int lane = threadIdx.x & 31;
  if (wid >= nrows) return;
  const float4 x4 = *(const float4*)(X + (size_t)wid * HD + lane * 4);
  float s = x4.x + x4.y + x4.z + x4.w;
  for (int o = 16; o > 0; o >>= 1) s += __shfl_xor(s, o, 32);
  const float mu = s * (1.f / HD);
  const float d0 = x4.x - mu, d1 = x4.y - mu, d2 = x4.z - mu, d3 = x4.w - mu;
  float vv = d0 * d0 + d1 * d1 + d2 * d2 + d3 * d3;
  for (int o = 16; o > 0; o >>= 1) vv += __shfl_xor(vv, o, 32);
  const float inv = rsqrtf(vv * (1.f / HD) + 1e-5f);
  const float4 g4 = *(const float4*)(gw + lane * 4);
  const float4 b4 = *(const float4*)(bw + lane * 4);
  float4 y;
  y.x = fmaxf(d0 * inv * g4.x + b4.x, 0.f);
  y.y = fmaxf(d1 * inv * g4.y + b4.y, 0.f);
  y.z = fmaxf(d2 * inv * g4.z + b4.z, 0.f);
  y.w = fmaxf(d3 * inv * g4.w + b4.w, 0.f);
  *(float4*)(Y + (size_t)wid * HD + lane * 4) = y;
}

// -----------------------------------------------------------------------------
// Per-block LDS partial sums per graph (batch is sorted -> low contention),
// then a handful of global atomics. Thread t owns column t for every graph.
// -----------------------------------------------------------------------------
__global__ __launch_bounds__(128) void pool_partial_kernel(
    const float* __restrict__ z, const int* __restrict__ batch,
    float* __restrict__ pool, float* __restrict__ cnt, int n)
{
  __shared__ float acc[NGRAPH * HD];
  __shared__ float cacc[NGRAPH];
  const int tid = threadIdx.x;   // column 0..127
  for (int g = 0; g < NGRAPH; ++g) acc[g * HD + tid] = 0.f;
  if (tid < NGRAPH) cacc[tid] = 0.f;
  __syncthreads();
  const int start = blockIdx.x * 256;
  const int end   = min(start + 256, n);
  for (int r = start; r < end; ++r) {
    const int g = batch[r];
    acc[g * HD + tid] += z[(size_t)r * HD + tid];
    if (tid == 0) cacc[g] += 1.f;
  }
  __syncthreads();
  for (int g = 0; g < NGRAPH; ++g) atomicAdd(&pool[g * HD + tid], acc[g * HD + tid]);
  if (tid < NGRAPH) atomicAdd(&cnt[tid], cacc[tid]);
}

// -----------------------------------------------------------------------------
// Single-block classifier: mean -> fc1 -> LN -> relu -> fc2 -> softmax,
// plus z_sel = z_pool[batch[0]]. 8 rows only; 8 waves handle LN per-row.
// -----------------------------------------------------------------------------
__global__ __launch_bounds__(256) void classifier_kernel(
    const float* __restrict__ pool, const float* __restrict__ cnt,
    const int* __restrict__ batch,
    const float* __restrict__ fc1_w, const float* __restrict__ fc1_b,
    const float* __restrict__ cg, const float* __restrict__ cb,
    const float* __restrict__ fc2_w, const float* __restrict__ fc2_b,
    float* __restrict__ probsOut, float* __restrict__ zselOut)
{
  __shared__ float mean[NGRAPH * HD];
  __shared__ float hc[NGRAPH * HD];
  __shared__ float lgt[NGRAPH * NCLASS];
  const int tid = threadIdx.x;
  for (int i = tid; i < NGRAPH * HD; i += 256) {
    const int g = i >> 7;
    mean[i] = pool[i] / fmaxf(cnt[g], 1.f);
  }
  __syncthreads();
  for (int i = tid; i < NGRAPH * HD; i += 256) {
    const int g = i >> 7, c = i & (HD - 1);
    float a = fc1_b[c];
    for (int kk = 0; kk < HD; ++kk) a += mean[g * HD + kk] * fc1_w[kk * HD + c];
    hc[i] = a;
  }
  __syncthreads();
  {
    const int wv = tid >> 5, ln = tid & 31;      // wave wv -> row wv (8 rows)
    float4 x4 = *(const float4*)(&hc[wv * HD + ln * 4]);
    float s = x4.x + x4.y + x4.z + x4.w;
    for (int o = 16; o > 0; o >>= 1) s += __shfl_xor(s, o, 32);
    const float mu = s * (1.f / HD);
    const float d0 = x4.x - mu, d1 = x4.y - mu, d2 = x4.z - mu, d3 = x4.w - mu;
    float vv = d0 * d0 + d1 * d1 + d2 * d2 + d3 * d3;
    for (int o = 16; o > 0; o >>= 1) vv += __shfl_xor(vv, o, 32);
    const float inv = rsqrtf(vv * (1.f / HD) + 1e-5f);
    const float4 g4 = *(const float4*)(cg + ln * 4);
    const float4 b4 = *(const float4*)(cb + ln * 4);
    float4 y;
    y.x = fmaxf(d0 * inv * g4.x + b4.x, 0.f);
    y.y = fmaxf(d1 * inv * g4.y + b4.y, 0.f);
    y.z = fmaxf(d2 * inv * g4.z + b4.z, 0.f);
    y.w = fmaxf(d3 * inv * g4.w + b4.w, 0.f);
    *(float4*)(&hc[wv * HD + ln * 4]) = y;
  }
  __syncthreads();
  if (tid < NGRAPH * NCLASS) {
    const int g = tid / NCLASS, c = tid % NCLASS;
    float a = fc2_b[c];
    for (int kk = 0; kk < HD; ++kk) a += hc[g * HD + kk] * fc2_w[kk * NCLASS + c];
    lgt[tid] = a;
  }
  __syncthreads();
  if (tid < NGRAPH) {
    float mx = -INFINITY;
    for (int c = 0; c < NCLASS; ++c) mx = fmaxf(mx, lgt[tid * NCLASS + c]);
    float ssum = 0.f, ex[NCLASS];
    for (int c = 0; c < NCLASS; ++c) { ex[c] = __expf(lgt[tid * NCLASS + c] - mx); ssum += ex[c]; }
    for (int c = 0; c < NCLASS; ++c) probsOut[tid * NCLASS + c] = ex[c] / ssum;
  }
  if (tid < HD) {
    const int g0 = batch[0];
    zselOut[tid] = mean[g0 * HD + tid];
  }
}

// -----------------------------------------------------------------------------
extern "C" void kernel_launch(void* const* d_in, const int* in_sizes, int n_in,
                              void* d_out, int out_size, void* d_ws, size_t ws_size,
                              hipStream_t stream) {
  (void)in_sizes; (void)n_in; (void)out_size; (void)ws_size;

  const float* x     = (const float*)d_in[0];
  const int*   ei    = (const int*)d_in[1];
  const float* attr  = (const float*)d_in[2];
  const int*   batch = (const int*)d_in[3];
  const int* src = ei;               // edge_index[0]
  const int* dst = ei + N_EDGES;     // edge_index[1]

  const float *Wq[3], *Wk[3], *Wv[3], *We[3], *Ws[3];
  const float *bq[3], *bk[3], *bv[3], *be[3], *bs[3];
  for (int l = 0; l < 3; ++l) {
    const int b = 4 + 10 * l;
    Wq[l] = (const float*)d_in[b + 0]; Wk[l] = (const float*)d_in[b + 1];
    Wv[l] = (const float*)d_in[b + 2]; We[l] = (const float*)d_in[b + 3];
    Ws[l] = (const float*)d_in[b + 4];
    bq[l] = (const float*)d_in[b + 5]; bk[l] = (const float*)d_in[b + 6];
    bv[l] = (const float*)d_in[b + 7]; be[l] = (const float*)d_in[b + 8];
    bs[l] = (const float*)d_in[b + 9];
  }
  const float* ln_g[2] = {(const float*)d_in[34], (const float*)d_in[36]};
  const float* ln_b[2] = {(const float*)d_in[35], (const float*)d_in[37]};
  const float* fc1_w = (const float*)d_in[38];
  const float* fc1_b = (const float*)d_in[39];
  const float* cln_g = (const float*)d_in[40];
  const float* cln_b = (const float*)d_in[41];
  const float* fc2_w = (const float*)d_in[42];
  const float* fc2_b = (const float*)d_in[43];

  // Workspace layout (~52 MB of f32)
  float* ws   = (float*)d_ws;
  float* qb   = ws;
  float* kb   = qb + (size_t)N_NODES * HD;
  float* vb   = kb + (size_t)N_NODES * HD;
  float* ob   = vb + (size_t)N_NODES * HD;
  float* hb   = ob + (size_t)N_NODES * HD;
  float* mb   = hb + (size_t)N_NODES * HD;
  float* sb   = mb + (size_t)N_NODES * NHEAD;
  float* pool = sb + (size_t)N_NODES * NHEAD;
  float* cnt  = pool + NGRAPH * HD;

  // Output layout: z, z_sel, probs, a1, a2, a3
  float* z     = (float*)d_out;
  float* zsel  = z + (size_t)N_NODES * HD;
  float* probs = zsel + HD;
  float* aout[3];
  aout[0] = probs + NGRAPH * NCLASS;
  aout[1] = aout[0] + (size_t)N_EDGES * NHEAD;
  aout[2] = aout[1] + (size_t)N_EDGES * NHEAD;

  const dim3 blk256(256);
  const dim3 gemmGrid(N_NODES / 16);                 // 1250 row tiles
  const int edgeWaveBlocks = (N_EDGES * 32) / 256;   // one wave per edge
  const int ehBlocks = (N_EDGES * NHEAD + 255) / 256;
  const int msBlocks = (N_NODES * NHEAD + 255) / 256;
  const int lnBlocks = (N_NODES * 32) / 256;

  auto run_layer = [&](const float* hin, int l, float* outBuf, float* aBuf) {
    gemm128_wmma_f32<<<gemmGrid, blk256, 0, stream>>>(hin, Wq[l], bq[l], qb, N_NODES);
    gemm128_wmma_f32<<<gemmGrid, blk256, 0, stream>>>(hin, Wk[l], bk[l], kb, N_NODES);
    gemm128_wmma_f32<<<gemmGrid, blk256, 0, stream>>>(hin, Wv[l], bv[l], vb, N_NODES);
    gemm128_wmma_f32<<<gemmGrid, blk256, 0, stream>>>(hin, Ws[l], bs[l], outBuf, N_NODES);
    fill_kernel<<<msBlocks, blk256, 0, stream>>>(mb, -INFINITY, N_NODES * NHEAD);
    fill_kernel<<<msBlocks, blk256, 0, stream>>>(sb, 0.f, N_NODES * NHEAD);
    edge_logits_kernel<<<edgeWaveBlocks, blk256, 0, stream>>>(
        qb, kb, src, dst, attr, We[l], be[l], aBuf, mb, N_EDGES);
    edge_exp_kernel<<<ehBlocks, blk256, 0, stream>>>(dst, aBuf, mb, sb, N_EDGES * NHEAD);
    edge_scatter_kernel<<<edgeWaveBlocks, blk256, 0, stream>>>(
        vb, src, dst, attr, We[l], be[l], aBuf, sb, outBuf, N_EDGES);
  };

  run_layer(x, 0, ob, aout[0]);
  ln_relu_kernel<<<lnBlocks, blk256, 0, stream>>>(ob, ln_g[0], ln_b[0], hb, N_NODES);
  run_layer(hb, 1, ob, aout[1]);
  ln_relu_kernel<<<lnBlocks, blk256, 0, stream>>>(ob, ln_g[1], ln_b[1], hb, N_NODES);
  run_layer(hb, 2, z, aout[2]);   // layer 3 writes z directly into d_out

  fill_kernel<<<(NGRAPH * HD + NGRAPH + 255) / 256, blk256, 0, stream>>>(
      pool, 0.f, NGRAPH * HD + NGRAPH);
  pool_partial_kernel<<<(N_NODES + 255) / 256, dim3(128), 0, stream>>>(
      z, batch, pool, cnt, N_NODES);
  classifier_kernel<<<1, blk256, 0, stream>>>(
      pool, cnt, batch, fc1_w, fc1_b, cln_g, cln_b, fc2_w, fc2_b, probs, zsel);
}